// GCNEncoder_77008763617309
// MI455X (gfx1250) — compile-verified
//
#include <hip/hip_runtime.h>
#include <hip/hip_bf16.h>

// GCN encoder for MI455X (gfx1250, wave32).
// Memory-bound on the edge gather/scatter (~3.5 GB vs 11 GFLOP), so GEMMs use
// full-precision V_WMMA_F32_16X16X4_F32 and sparse propagate uses wave-per-edge
// float4 gathers + hardware f32 global atomics.
// GEMM blocks cover 32 rows so each B fragment feeds two WMMAs (halves W
// re-streaming from L2).

typedef __attribute__((ext_vector_type(2))) float v2f;
typedef __attribute__((ext_vector_type(8))) float v8f;

#define IN_CH 256
#define HID_CH 128
#define GEMM_ROWS 32   // rows per block (N % 32 == 0)

// ---------------- degree / normalization ----------------

__global__ __launch_bounds__(256) void k_init_deg(float* deg, int N) {
    int i = blockIdx.x * 256 + threadIdx.x;
    if (i < N) deg[i] = 1.0f;  // self-loop weight 1.0
}

__global__ __launch_bounds__(256) void k_deg_scatter(const int* __restrict__ col,
                                                     const float* __restrict__ w,
                                                     float* deg, int E) {
    int e = blockIdx.x * 256 + threadIdx.x;
    if (e < E) unsafeAtomicAdd(&deg[col[e]], w[e]);
}

__global__ __launch_bounds__(256) void k_dinv(const float* __restrict__ deg,
                                              float* dinv, int N) {
    int i = blockIdx.x * 256 + threadIdx.x;
    if (i < N) {
        float d = deg[i];
        dinv[i] = (d > 0.0f) ? rsqrtf(d) : 0.0f;
    }
}

__global__ __launch_bounds__(256) void k_edge_norm(const int* __restrict__ row,
                                                   const int* __restrict__ col,
                                                   const float* __restrict__ w,
                                                   const float* __restrict__ dinv,
                                                   float* nrm, int E) {
    int e = blockIdx.x * 256 + threadIdx.x;
    if (e < E) nrm[e] = dinv[row[e]] * w[e] * dinv[col[e]];
}

// ---------------- dense GEMM: H[N x 128] = A[N x K] * W[K x 128] ------------
// Block = 32 rows x 128 cols; 8 waves; each wave owns a 16-wide column tile
// and two 16x16 accumulators (row tiles), sharing one B fragment per k-step.
// A staged in LDS with stride K+4: fragment-read banks are 4*L mod 64,
// distinct for L<16 in each half-wave -> conflict-free.

__global__ __launch_bounds__(256) void k_gemm_wmma(const float* __restrict__ A,
                                                   const float* __restrict__ W,
                                                   float* __restrict__ H, int K) {
    __shared__ float As[GEMM_ROWS * (IN_CH + 4)];
    const int stride = K + 4;
    const int tid = threadIdx.x;
    const int rowbase = blockIdx.x * GEMM_ROWS;

    for (int idx = tid; idx < GEMM_ROWS * K; idx += 256) {
        int r = idx / K;
        int kk = idx - r * K;
        As[r * stride + kk] = A[(size_t)(rowbase + r) * K + kk];
    }
    __syncthreads();

    const int wave = tid >> 5;
    const int lane = tid & 31;
    const int m    = lane & 15;   // row (A frag) / col (B,D frags) within tile
    const int half = lane >> 4;   // lane-half selects K sub-block
    const int cb   = wave * 16;   // this wave's column tile

    v8f c0 = {};
    v8f c1 = {};
    const float* wp  = W + cb + m;                   // column m of W tile
    const float* ap0 = As + m * stride;              // row tile 0
    const float* ap1 = As + (16 + m) * stride;       // row tile 1

    for (int k = 0; k < K; k += 4) {
        const int koff = k + half * 2;
        v2f b;
        b[0] = wp[(size_t)koff * HID_CH];
        b[1] = wp[(size_t)(koff + 1) * HID_CH];
        v2f a0, a1;
        a0[0] = ap0[koff];
        a0[1] = ap0[koff + 1];
        a1[0] = ap1[koff];
        a1[1] = ap1[koff + 1];
        c0 = __builtin_amdgcn_wmma_f32_16x16x4_f32(false, a0, false, b,
                                                   (short)0, c0, false, false);
        c1 = __builtin_amdgcn_wmma_f32_16x16x4_f32(false, a1, false, b,
                                                   (short)0, c1, false, false);
    }

    // D layout: VGPR r, lane L -> (M = r + 8*(L/16), N = L%16)
    float* hp0 = H + (size_t)rowbase * HID_CH + cb + m;
    float* hp1 = hp0 + (size_t)16 * HID_CH;
#pragma unroll
    for (int r = 0; r < 8; ++r) {
        hp0[(size_t)(r + 8 * half) * HID_CH] = c0[r];
        hp1[(size_t)(r + 8 * half) * HID_CH] = c1[r];
    }
}

// ---------------- sparse propagate ----------------

// acc[i] = dinv[i]^2 * h[i]  (self-loop contribution; also initializes acc)
__global__ __launch_bounds__(256) void k_init_acc(const float* __restrict__ h,
                                                  const float* __restrict__ dinv,
                                                  float* __restrict__ acc, int N) {
    int t = blockIdx.x * 256 + threadIdx.x;   // N*32 threads
    int node = t >> 5, chunk = t & 31;
    if (node >= N) return;
    float s = dinv[node] * dinv[node];
    const float4* hp = (const float4*)(h + (size_t)node * HID_CH);
    float4* ap = (float4*)(acc + (size_t)node * HID_CH);
    float4 v = hp[chunk];
    v.x *= s; v.y *= s; v.z *= s; v.w *= s;
    ap[chunk] = v;
}

// wave32 per edge: acc[col] += nrm * h[row]
__global__ __launch_bounds__(256) void k_scatter(const int* __restrict__ row,
                                                 const int* __restrict__ col,
                                                 const float* __restrict__ nrm,
                                                 const float* __restrict__ h,
                                                 float* __restrict__ acc, int E) {
    int wid = (blockIdx.x * 256 + threadIdx.x) >> 5;
    int lane = threadIdx.x & 31;
    if (wid >= E) return;
    int r = row[wid];
    int c = col[wid];
    float nv = nrm[wid];
    const float4* hp = (const float4*)(h + (size_t)r * HID_CH);
    float4 v = hp[lane];
    float* dst = acc + (size_t)c * HID_CH + lane * 4;
    unsafeAtomicAdd(dst + 0, v.x * nv);
    unsafeAtomicAdd(dst + 1, v.y * nv);
    unsafeAtomicAdd(dst + 2, v.z * nv);
    unsafeAtomicAdd(dst + 3, v.w * nv);
}

// acc = relu(acc + b), in place
__global__ __launch_bounds__(256) void k_bias_relu(float* __restrict__ acc,
                                                   const float* __restrict__ b, int N) {
    int g = blockIdx.x * 256 + threadIdx.x;   // N*32 float4s
    if (g >= N * 32) return;
    float4* ap = (float4*)acc;
    const float4* bp = (const float4*)b;
    float4 v = ap[g];
    float4 bb = bp[g & 31];
    v.x = fmaxf(v.x + bb.x, 0.0f);
    v.y = fmaxf(v.y + bb.y, 0.0f);
    v.z = fmaxf(v.z + bb.z, 0.0f);
    v.w = fmaxf(v.w + bb.w, 0.0f);
    ap[g] = v;
}

// out = (out + b) / max(||out + b||_2, 1e-12), wave32 per node
__global__ __launch_bounds__(256) void k_bias_l2norm(float* __restrict__ out,
                                                     const float* __restrict__ b, int N) {
    int wid = (blockIdx.x * 256 + threadIdx.x) >> 5;
    int lane = threadIdx.x & 31;
    if (wid >= N) return;
    float4* op = (float4*)(out + (size_t)wid * HID_CH);
    const float4* bp = (const float4*)b;
    float4 v = op[lane];
    float4 bb = bp[lane];
    v.x += bb.x; v.y += bb.y; v.z += bb.z; v.w += bb.w;
    float ss = v.x * v.x + v.y * v.y + v.z * v.z + v.w * v.w;
#pragma unroll
    for (int m = 16; m >= 1; m >>= 1) ss += __shfl_xor(ss, m, 32);
    float inv = 1.0f / fmaxf(sqrtf(ss), 1e-12f);
    v.x *= inv; v.y *= inv; v.z *= inv; v.w *= inv;
    op[lane] = v;
}

// ---------------- launcher ----------------

static inline size_t align_up(size_t v, size_t a) { return (v + a - 1) & ~(a - 1); }

extern "C" void kernel_launch(void* const* d_in, const int* in_sizes, int n_in,
                              void* d_out, int out_size, void* d_ws, size_t ws_size,
                              hipStream_t stream) {
    const float* x  = (const float*)d_in[0];
    const int*   ei = (const int*)d_in[1];
    const float* ew = (const float*)d_in[2];
    const float* W1 = (const float*)d_in[3];
    const float* b1 = (const float*)d_in[4];
    const float* W2 = (const float*)d_in[5];
    const float* b2 = (const float*)d_in[6];
    float* out = (float*)d_out;

    const int N = in_sizes[0] / IN_CH;   // 100000
    const int E = in_sizes[1] / 2;       // 1600000
    const int* row = ei;
    const int* col = ei + E;

    // workspace layout (bytes, 1 KiB aligned)
    char* ws = (char*)d_ws;
    size_t off = 0;
    float* deg  = (float*)(ws + off); off = align_up(off + (size_t)N * 4, 1024);
    float* dinv = (float*)(ws + off); off = align_up(off + (size_t)N * 4, 1024);
    float* nrm  = (float*)(ws + off); off = align_up(off + (size_t)E * 4, 1024);
    float* h    = (float*)(ws + off); off = align_up(off + (size_t)N * HID_CH * 4, 1024);
    float* acc  = (float*)(ws + off); off = align_up(off + (size_t)N * HID_CH * 4, 1024);
    (void)ws_size; (void)n_in; (void)out_size;

    const int nb_nodes  = (N + 255) / 256;        // thread-per-node
    const int nb_edges  = (E + 255) / 256;        // thread-per-edge
    const int nb_nvec   = (N * 32 + 255) / 256;   // thread-per-float4 (N*32)
    const int nb_nwave  = (N + 7) / 8;            // wave-per-node
    const int nb_ewave  = (E + 7) / 8;            // wave-per-edge
    const int nb_gemm   = N / GEMM_ROWS;          // 32-row tiles (N % 32 == 0)

    // normalization coefficients
    k_init_deg   <<<nb_nodes, 256, 0, stream>>>(deg, N);
    k_deg_scatter<<<nb_edges, 256, 0, stream>>>(col, ew, deg, E);
    k_dinv       <<<nb_nodes, 256, 0, stream>>>(deg, dinv, N);
    k_edge_norm  <<<nb_edges, 256, 0, stream>>>(row, col, ew, dinv, nrm, E);

    // layer 1: h = x @ W1 ; acc = self + scatter ; relu(acc + b1)
    k_gemm_wmma  <<<nb_gemm, 256, 0, stream>>>(x, W1, h, IN_CH);
    k_init_acc   <<<nb_nvec, 256, 0, stream>>>(h, dinv, acc, N);
    k_scatter    <<<nb_ewave, 256, 0, stream>>>(row, col, nrm, h, acc, E);
    k_bias_relu  <<<nb_nvec, 256, 0, stream>>>(acc, b1, N);

    // layer 2: h = acc @ W2 ; out = self + scatter ; (out + b2) L2-normalized
    k_gemm_wmma  <<<nb_gemm, 256, 0, stream>>>(acc, W2, h, HID_CH);
    k_init_acc   <<<nb_nvec, 256, 0, stream>>>(h, dinv, out, N);
    k_scatter    <<<nb_ewave, 256, 0, stream>>>(row, col, nrm, h, out, E);
    k_bias_l2norm<<<nb_nwave, 256, 0, stream>>>(out, b2, N);
}